// KANActivation_31559419691078
// MI455X (gfx1250) — compile-verified
//
#include <hip/hip_runtime.h>

// KAN activation: out = wb * silu(x) + ws * BSpline(knots, coeffs, k=3)(x)
// x: (8,4096,1024) fp32. Memory-bound streaming kernel (~268 MB/call,
// ~11.5 us floor at 23.3 TB/s).
//
//  1) setup kernel (1 thread): symbolic de Boor -> exact monomial cubic per
//     clipped interval i in {3,4,5}; coefficients pre-scaled by ws. d_ws:
//       ws[0..3]=ws*poly(i=3), ws[4..7]=ws*poly(i=4), ws[8..11]=ws*poly(i=5),
//       ws[12]=t[4], ws[13]=t[5], ws[14]=wb, ws[15]=ws
//  2) streaming kernel: x2-unrolled grid-stride, float4 NT loads/stores,
//     branchless 2-level cndmask interval select, Horner (3 FMA), silu via
//     v_exp_f32 + v_rcp_f32, global_prefetch_b8 ahead of the stream.

typedef float v4f __attribute__((ext_vector_type(4)));

#define KN 10
#define KK 3

__global__ void kan_setup_kernel(const float* __restrict__ knots,
                                 const float* __restrict__ coeffs,
                                 const float* __restrict__ wb,
                                 const float* __restrict__ wsw,
                                 float* __restrict__ params) {
  if (threadIdx.x != 0 || blockIdx.x != 0) return;
  float t[KN];
  for (int j = 0; j < KN; ++j) t[j] = knots[j];
  float c[6];
  for (int j = 0; j < 6; ++j) c[j] = coeffs[j];
  const float wsv = wsw[0];

  // For each clipped interval i in {3,4,5}, run de Boor on polynomial
  // coefficient vectors (d[j] is a poly in x, alpha = A*x + B is linear).
  for (int i = 3; i <= 5; ++i) {
    float d[4][4];
    for (int j = 0; j < 4; ++j) {
      d[j][0] = c[i - KK + j];
      d[j][1] = 0.0f; d[j][2] = 0.0f; d[j][3] = 0.0f;
    }
    for (int r = 1; r <= KK; ++r) {
      for (int j = KK; j >= r; --j) {
        float t_lo = t[i + j - KK];
        float t_hi = t[i + j + 1 - r];
        float inv  = __builtin_amdgcn_rcpf(t_hi - t_lo);
        float A = inv, B = -t_lo * inv;           // alpha(x) = A*x + B
        // d[j] = d[j-1] + alpha * (d[j] - d[j-1])
        float e[4], nd[4];
        for (int m = 0; m < 4; ++m) e[m] = d[j][m] - d[j - 1][m];
        for (int m = 0; m < 4; ++m)
          nd[m] = d[j - 1][m] + B * e[m] + (m > 0 ? A * e[m - 1] : 0.0f);
        for (int m = 0; m < 4; ++m) d[j][m] = nd[m];
      }
    }
    for (int m = 0; m < 4; ++m) params[(i - 3) * 4 + m] = wsv * d[KK][m];
  }
  params[12] = t[4];   // i=3 vs 4 threshold (searchsorted 'right' -> x >= t4)
  params[13] = t[5];   // i=4 vs 5 threshold
  params[14] = wb[0];
  params[15] = wsv;
}

__device__ __forceinline__ float kan_eval(
    float xv,
    float a0, float a1, float a2, float a3,
    float b0, float b1, float b2, float b3,
    float c0, float c1, float c2, float c3,
    float t4, float t5, float wbv) {
  bool s1 = xv >= t4;
  bool s2 = xv >= t5;
  float p0 = s2 ? c0 : (s1 ? b0 : a0);
  float p1 = s2 ? c1 : (s1 ? b1 : a1);
  float p2 = s2 ? c2 : (s1 ? b2 : a2);
  float p3 = s2 ? c3 : (s1 ? b3 : a3);
  // poly already scaled by ws at setup time
  float sp = __builtin_fmaf(__builtin_fmaf(__builtin_fmaf(p3, xv, p2), xv, p1), xv, p0);
  // silu(x) = x / (1 + exp(-x));  exp(-x) = exp2(-x * log2(e))
  float e   = __builtin_amdgcn_exp2f(xv * -1.4426950408889634f);
  float sig = __builtin_amdgcn_rcpf(1.0f + e);
  return __builtin_fmaf(wbv, xv * sig, sp);
}

__global__ void __launch_bounds__(256)
kan_main_kernel(const float* __restrict__ x,
                const float* __restrict__ params,
                float* __restrict__ out,
                long long n) {
  __shared__ float p[16];
  if (threadIdx.x < 16) p[threadIdx.x] = params[threadIdx.x];
  __syncthreads();
  const float a0 = p[0],  a1 = p[1],  a2 = p[2],  a3 = p[3];
  const float b0 = p[4],  b1 = p[5],  b2 = p[6],  b3 = p[7];
  const float c0 = p[8],  c1 = p[9],  c2 = p[10], c3 = p[11];
  const float t4 = p[12], t5 = p[13], wbv = p[14];

  const long long n4     = n >> 2;
  const long long tid    = (long long)blockIdx.x * blockDim.x + threadIdx.x;
  const long long stride = (long long)gridDim.x * blockDim.x;

  const v4f* __restrict__ xi = (const v4f*)x;
  v4f* __restrict__ oo       = (v4f*)out;

  long long i = tid;
  // x2-unrolled main loop: both B128 loads in flight before compute
  for (; i + stride < n4; i += 2 * stride) {
    long long pf = i + 16 * stride;
    if (pf < n4) __builtin_prefetch(&xi[pf], 0, 0);            // global_prefetch_b8
    v4f v0 = __builtin_nontemporal_load(&xi[i]);
    v4f v1 = __builtin_nontemporal_load(&xi[i + stride]);
    v4f r0, r1;
    r0[0] = kan_eval(v0[0], a0,a1,a2,a3, b0,b1,b2,b3, c0,c1,c2,c3, t4,t5,wbv);
    r0[1] = kan_eval(v0[1], a0,a1,a2,a3, b0,b1,b2,b3, c0,c1,c2,c3, t4,t5,wbv);
    r0[2] = kan_eval(v0[2], a0,a1,a2,a3, b0,b1,b2,b3, c0,c1,c2,c3, t4,t5,wbv);
    r0[3] = kan_eval(v0[3], a0,a1,a2,a3, b0,b1,b2,b3, c0,c1,c2,c3, t4,t5,wbv);
    r1[0] = kan_eval(v1[0], a0,a1,a2,a3, b0,b1,b2,b3, c0,c1,c2,c3, t4,t5,wbv);
    r1[1] = kan_eval(v1[1], a0,a1,a2,a3, b0,b1,b2,b3, c0,c1,c2,c3, t4,t5,wbv);
    r1[2] = kan_eval(v1[2], a0,a1,a2,a3, b0,b1,b2,b3, c0,c1,c2,c3, t4,t5,wbv);
    r1[3] = kan_eval(v1[3], a0,a1,a2,a3, b0,b1,b2,b3, c0,c1,c2,c3, t4,t5,wbv);
    __builtin_nontemporal_store(r0, &oo[i]);
    __builtin_nontemporal_store(r1, &oo[i + stride]);
  }
  if (i < n4) {
    v4f v = __builtin_nontemporal_load(&xi[i]);
    v4f r;
    r[0] = kan_eval(v[0], a0,a1,a2,a3, b0,b1,b2,b3, c0,c1,c2,c3, t4,t5,wbv);
    r[1] = kan_eval(v[1], a0,a1,a2,a3, b0,b1,b2,b3, c0,c1,c2,c3, t4,t5,wbv);
    r[2] = kan_eval(v[2], a0,a1,a2,a3, b0,b1,b2,b3, c0,c1,c2,c3, t4,t5,wbv);
    r[3] = kan_eval(v[3], a0,a1,a2,a3, b0,b1,b2,b3, c0,c1,c2,c3, t4,t5,wbv);
    __builtin_nontemporal_store(r, &oo[i]);
  }
  // scalar tail (n not divisible by 4)
  for (long long j = (n4 << 2) + tid; j < n; j += stride) {
    float v = x[j];
    out[j] = kan_eval(v, a0,a1,a2,a3, b0,b1,b2,b3, c0,c1,c2,c3, t4,t5,wbv);
  }
}

extern "C" void kernel_launch(void* const* d_in, const int* in_sizes, int n_in,
                              void* d_out, int out_size, void* d_ws, size_t ws_size,
                              hipStream_t stream) {
  const float* x      = (const float*)d_in[0];
  const float* knots  = (const float*)d_in[1];
  const float* coeffs = (const float*)d_in[2];
  const float* wb     = (const float*)d_in[3];
  const float* wsw    = (const float*)d_in[4];
  float* out    = (float*)d_out;
  float* params = (float*)d_ws;   // 16 floats

  const long long n = (long long)out_size;

  kan_setup_kernel<<<1, 32, 0, stream>>>(knots, coeffs, wb, wsw, params);

  const int threads = 256;                 // 8 wave32s per block
  long long n4 = n >> 2;
  long long want = (n4 + threads - 1) / threads;
  int blocks = (int)(want < 4096 ? (want > 0 ? want : 1) : 4096);
  kan_main_kernel<<<blocks, threads, 0, stream>>>(x, params, out, n);
}